// EdgeMaskNet_81226421502637
// MI455X (gfx1250) — compile-verified
//
#include <hip/hip_runtime.h>
#include <hip/hip_bf16.h>

typedef float v2f __attribute__((ext_vector_type(2)));
typedef float v8f __attribute__((ext_vector_type(8)));

#define N_NODES 50000
#define E_EDGES 800000
#define IN_DIMC 128
#define HIDC    64
#define KFAC    4
#define ZCOLS   256          // KFAC * HIDC
#define W1_PITCH 160         // words per packed K-pair row: 32 mod 64 -> half-wave bank skew

// xor-swizzle within 16-lane groups (masks < 16 never cross the half boundary)
#define SWZ_XOR(v, m) \
  __int_as_float(__builtin_amdgcn_ds_swizzle(__float_as_int(v), ((m) << 10) | 31))

// ---------------------------------------------------------------- init
__global__ void init_kernel(float* __restrict__ disen) {
  int t = threadIdx.x;
  if (t < ZCOLS) disen[t] = 0.0f;   // Z >= 0 post-relu, so 0 is a safe max-identity
}

// ---------------------------------------------------------------- Z = normalize(relu(x @ W'))
// 128 threads (4 waves), 64 rows/block. Wave w owns row-tile w (16 rows).
__global__ __launch_bounds__(128) void z_kernel(const float* __restrict__ x,
                                                const float* __restrict__ Wk,   // (K,128,64)
                                                const float* __restrict__ bk,   // (K,64)
                                                float* __restrict__ Zg,         // (N,256)
                                                float* __restrict__ disen) {    // (256)
  __shared__ __align__(16) float xs[64 * IN_DIMC];   // 32 KB staged x rows

  const int tid  = threadIdx.x;
  const int lane = tid & 31;
  const int w    = tid >> 5;
  const int row0 = blockIdx.x * 64;
  const bool fullblk = (row0 + 64 <= N_NODES);       // uniform: 781 of 782 blocks

  // stage x via async global->LDS DMA (B128); zero-fill rows past N with plain DS stores
  for (int i = tid; i < 64 * IN_DIMC / 4; i += 128) {
    int r = i >> 5, c4 = (i & 31) * 4;
    int grow = row0 + r;
    if (grow < N_NODES) {
      unsigned int lds  = (unsigned int)(uintptr_t)(xs + r * IN_DIMC + c4);
      unsigned long long gp = (unsigned long long)(uintptr_t)(x + (size_t)grow * IN_DIMC + c4);
      asm volatile("global_load_async_to_lds_b128 %0, %1, off"
                   :: "v"(lds), "v"(gp) : "memory");
    } else {
      *(float4*)(xs + r * IN_DIMC + c4) = make_float4(0.f, 0.f, 0.f, 0.f);
    }
  }
  asm volatile("s_wait_asynccnt 0x0" ::: "memory");
  __syncthreads();

  const int half  = lane >> 4;
  const int idx16 = lane & 15;
  const int rt    = w;

  for (int ct = 0; ct < 16; ++ct) {        // 16 col tiles -> 256 cols
    v8f acc = {};
    const int col = ct * 16 + idx16;
    const int kk  = col >> 6;
    const int hh  = col & 63;
    #pragma unroll 4
    for (int kc = 0; kc < 32; ++kc) {      // K = 128 in chunks of 4
      const int kb = kc * 4 + half * 2;    // A: lanes0-15 K={0,1}, lanes16-31 K={2,3}
      v2f a = *(const v2f*)(xs + (rt * 16 + idx16) * IN_DIMC + kb);
      v2f b;
      b.x = Wk[(size_t)kk * (IN_DIMC * HIDC) + kb * HIDC + hh];
      b.y = Wk[(size_t)kk * (IN_DIMC * HIDC) + (kb + 1) * HIDC + hh];
      acc = __builtin_amdgcn_wmma_f32_16x16x4_f32(false, a, false, b,
                                                  (short)0, acc, false, false);
    }
    const float bias = bk[col];
    if (fullblk) {                          // common path: no per-row exec juggling
      #pragma unroll
      for (int r = 0; r < 8; ++r) {
        int grow = row0 + rt * 16 + r + half * 8;
        Zg[(size_t)grow * ZCOLS + col] = fmaxf(acc[r] + bias, 0.0f);
      }
    } else {
      #pragma unroll
      for (int r = 0; r < 8; ++r) {
        int grow = row0 + rt * 16 + r + half * 8;
        if (grow < N_NODES)
          Zg[(size_t)grow * ZCOLS + col] = fmaxf(acc[r] + bias, 0.0f);
      }
    }
  }
  __threadfence();
  __syncthreads();

  // normalize: each thread owns whole (row,k) 64-vectors (two passes through L1-hot data)
  for (int p = tid; p < 64 * KFAC; p += 128) {
    int lr = p >> 2, k = p & 3;
    int grow = row0 + lr;
    if (grow >= N_NODES) continue;
    float4* zp = (float4*)(Zg + (size_t)grow * ZCOLS + k * HIDC);
    float s = 0.f;
    #pragma unroll 4
    for (int h4 = 0; h4 < 16; ++h4) {
      float4 q = zp[h4];
      s += q.x * q.x + q.y * q.y + q.z * q.z + q.w * q.w;
    }
    float scale = 1.0f / fmaxf(sqrtf(s), 1e-12f);
    #pragma unroll 4
    for (int h4 = 0; h4 < 16; ++h4) {
      float4 q = zp[h4];
      q.x *= scale; q.y *= scale; q.z *= scale; q.w *= scale;
      zp[h4] = q;
    }
  }
  __threadfence();
  __syncthreads();

  // column max: thread t owns columns t, t+128; adjacent lanes -> coalesced L1-hot reads
  const int rmax = fullblk ? 64 : (N_NODES - row0);
  for (int c = tid; c < ZCOLS; c += 128) {
    float m = 0.f;
    for (int lr = 0; lr < rmax; ++lr)
      m = fmaxf(m, Zg[(size_t)(row0 + lr) * ZCOLS + c]);
    atomicMax((int*)disen + c, __float_as_int(m));   // >=0 -> int bits order == float order
  }
}

// ---------------------------------------------------------------- edge MLP
// 128 threads (4 waves). Wave w handles 16 edges across all k.
// W1 repacked in LDS: w1t[p*160 + o*2 + j], p = K-pair row (0..31 src, 32..63 dst), j = K parity.
// -> one conflict-free ds_load_b64 per B operand (pitch 160 = 32 mod 64 banks skews the halves).
__global__ __launch_bounds__(128) void edge_kernel(const int* __restrict__ ei,   // (2,E)
                                                   const float* __restrict__ Zg, // (N,256)
                                                   const float* __restrict__ W1, // (128,64)
                                                   const float* __restrict__ b1, // (64)
                                                   const float* __restrict__ W2, // (64,1)
                                                   const float* __restrict__ b2, // (1)
                                                   float* __restrict__ e_final,  // (E)
                                                   float* __restrict__ e_all) {  // (K,E)
  __shared__ __align__(16) float w1t[64 * W1_PITCH];   // 40 KB packed W1
  const int tid  = threadIdx.x;
  const int lane = tid & 31;
  const int w    = tid >> 5;

  for (int i = tid; i < 2 * HIDC * HIDC; i += 128) {   // coalesced read, packed scatter
    int row = i >> 6;                  // 0..127 (0..63 = W1s, 64..127 = W1d)
    int o   = i & 63;
    w1t[(row >> 1) * W1_PITCH + o * 2 + (row & 1)] = W1[i];
  }
  __syncthreads();

  const int half  = lane >> 4;
  const int idx16 = lane & 15;
  const int ebase = blockIdx.x * 64 + w * 16;          // E % 64 == 0
  const int myE   = ebase + idx16;
  const int srow  = ei[myE];
  const int drow  = ei[E_EDGES + myE];
  const float* zs = Zg + (size_t)srow * ZCOLS;
  const float* zd = Zg + (size_t)drow * ZCOLS;

  float b1v[4], w2v[4];
  #pragma unroll
  for (int t = 0; t < 4; ++t) {
    b1v[t] = b1[t * 16 + idx16];
    w2v[t] = W2[t * 16 + idx16];
  }
  const float b2v = b2[0];

  // butterfly lands e for row (lane&7)+8*half on every lane; lanes 0-7 / 16-23 own writes
  const int  ownerM = half * 8 + (lane & 7);
  const bool owner  = (lane & 8) == 0;
  const bool s0 = lane & 1, s1 = lane & 2, s2 = lane & 4;
  float emax = -INFINITY;

  for (int k = 0; k < KFAC; ++k) {
    v8f acc0 = {}, acc1 = {}, acc2 = {}, acc3 = {};
    #pragma unroll
    for (int kc = 0; kc < 16; ++kc) {                  // K = 64 in chunks of 4
      const int kb = kc * 4 + half * 2;
      const int ps = (kc * 2 + half) * W1_PITCH;       // src K-pair row
      const int pd = (32 + kc * 2 + half) * W1_PITCH;  // dst K-pair row
      v2f as = *(const v2f*)(zs + k * HIDC + kb);
      v2f ad = *(const v2f*)(zd + k * HIDC + kb);
      v2f bs0 = *(const v2f*)(w1t + ps + (0 * 16 + idx16) * 2);
      v2f bs1 = *(const v2f*)(w1t + ps + (1 * 16 + idx16) * 2);
      v2f bs2 = *(const v2f*)(w1t + ps + (2 * 16 + idx16) * 2);
      v2f bs3 = *(const v2f*)(w1t + ps + (3 * 16 + idx16) * 2);
      acc0 = __builtin_amdgcn_wmma_f32_16x16x4_f32(false, as, false, bs0, (short)0, acc0, false, false);
      acc1 = __builtin_amdgcn_wmma_f32_16x16x4_f32(false, as, false, bs1, (short)0, acc1, false, false);
      acc2 = __builtin_amdgcn_wmma_f32_16x16x4_f32(false, as, false, bs2, (short)0, acc2, false, false);
      acc3 = __builtin_amdgcn_wmma_f32_16x16x4_f32(false, as, false, bs3, (short)0, acc3, false, false);
      v2f bd0 = *(const v2f*)(w1t + pd + (0 * 16 + idx16) * 2);
      v2f bd1 = *(const v2f*)(w1t + pd + (1 * 16 + idx16) * 2);
      v2f bd2 = *(const v2f*)(w1t + pd + (2 * 16 + idx16) * 2);
      v2f bd3 = *(const v2f*)(w1t + pd + (3 * 16 + idx16) * 2);
      acc0 = __builtin_amdgcn_wmma_f32_16x16x4_f32(false, ad, false, bd0, (short)0, acc0, false, false);
      acc1 = __builtin_amdgcn_wmma_f32_16x16x4_f32(false, ad, false, bd1, (short)0, acc1, false, false);
      acc2 = __builtin_amdgcn_wmma_f32_16x16x4_f32(false, ad, false, bd2, (short)0, acc2, false, false);
      acc3 = __builtin_amdgcn_wmma_f32_16x16x4_f32(false, ad, false, bd3, (short)0, acc3, false, false);
    }
    // per-lane partials of e[M] = sum_o relu(C[M,o] + b1[o]) * W2[o]
    float er[8];
    #pragma unroll
    for (int r = 0; r < 8; ++r) {
      er[r] = fmaxf(acc0[r] + b1v[0], 0.0f) * w2v[0]
            + fmaxf(acc1[r] + b1v[1], 0.0f) * w2v[1]
            + fmaxf(acc2[r] + b1v[2], 0.0f) * w2v[2]
            + fmaxf(acc3[r] + b1v[3], 0.0f) * w2v[3];
    }
    // 8-value butterfly transpose-reduce over 16 lanes (8 swizzles vs 32 bpermutes)
    float t, p0, p1, p2, p3, q0, q1, r0;
    t = s0 ? er[0] : er[1];  p0 = (s0 ? er[1] : er[0]) + SWZ_XOR(t, 1);
    t = s0 ? er[2] : er[3];  p1 = (s0 ? er[3] : er[2]) + SWZ_XOR(t, 1);
    t = s0 ? er[4] : er[5];  p2 = (s0 ? er[5] : er[4]) + SWZ_XOR(t, 1);
    t = s0 ? er[6] : er[7];  p3 = (s0 ? er[7] : er[6]) + SWZ_XOR(t, 1);
    t = s1 ? p0 : p1;        q0 = (s1 ? p1 : p0) + SWZ_XOR(t, 2);
    t = s1 ? p2 : p3;        q1 = (s1 ? p3 : p2) + SWZ_XOR(t, 2);
    t = s2 ? q0 : q1;        r0 = (s2 ? q1 : q0) + SWZ_XOR(t, 4);
    float ev = r0 + SWZ_XOR(r0, 8) + b2v;   // lane l now holds e for M = (l&7)+8*half
    if (owner) {
      e_all[(size_t)k * E_EDGES + ebase + ownerM] = ev;
      emax = fmaxf(emax, ev);
    }
  }
  if (owner) e_final[ebase + ownerM] = emax;
}

// ---------------------------------------------------------------- launch
extern "C" void kernel_launch(void* const* d_in, const int* in_sizes, int n_in,
                              void* d_out, int out_size, void* d_ws, size_t ws_size,
                              hipStream_t stream) {
  const float* x  = (const float*)d_in[0];
  const int*   ei = (const int*)d_in[1];
  const float* Wk = (const float*)d_in[2];
  const float* bk = (const float*)d_in[3];
  const float* W1 = (const float*)d_in[4];
  const float* b1 = (const float*)d_in[5];
  const float* W2 = (const float*)d_in[6];
  const float* b2 = (const float*)d_in[7];

  float* out     = (float*)d_out;
  float* disen   = out;                                  // 256
  float* Zg      = out + ZCOLS;                          // N*256
  float* e_final = Zg + (size_t)N_NODES * ZCOLS;         // E
  float* e_all   = e_final + E_EDGES;                    // K*E

  hipLaunchKernelGGL(init_kernel, dim3(1), dim3(256), 0, stream, disen);
  hipLaunchKernelGGL(z_kernel, dim3((N_NODES + 63) / 64), dim3(128), 0, stream,
                     x, Wk, bk, Zg, disen);
  hipLaunchKernelGGL(edge_kernel, dim3(E_EDGES / 64), dim3(128), 0, stream,
                     ei, Zg, W1, b1, W2, b2, e_final, e_all);
}